// ContextSeg_17051020165525
// MI455X (gfx1250) — compile-verified
//
#include <hip/hip_runtime.h>
#include <hip/hip_bf16.h>
#include <cstddef>
#include <cstdint>

typedef __attribute__((ext_vector_type(2))) float v2f;
typedef __attribute__((ext_vector_type(4))) float v4f;
typedef __attribute__((ext_vector_type(8))) float v8f;

#define BDIM 4
#define CDIM 8
#define HW64 4096     // 64*64
#define SDIM 512
#define QDIM 4096     // (512/8)^2
#define KDIM 256      // attn K / unfold L
#define PSQ 64        // 8*8 patch elements
#define LDA 36        // padded LDS row stride (floats): 16B-aligned, conflict-free

// ---------------------------------------------------------------------------
// CDNA5 async global->LDS helpers (ASYNCcnt-tracked DMA, no VGPR staging).
// lds: byte address within group segment (low 32 bits of the flat pointer).
// ---------------------------------------------------------------------------
__device__ __forceinline__ void async_ld16_nt(unsigned lds, const float* g)
{
    asm volatile("global_load_async_to_lds_b128 %0, %1, off th:TH_LOAD_NT"
                 :: "v"(lds), "v"(g) : "memory");
}
__device__ __forceinline__ void async_ld16(unsigned lds, const float* g)
{
    asm volatile("global_load_async_to_lds_b128 %0, %1, off"
                 :: "v"(lds), "v"(g) : "memory");
}
__device__ __forceinline__ void wait_async6()
{
    asm volatile("s_wait_asynccnt 0x6" ::: "memory");
}
__device__ __forceinline__ void wait_async0()
{
    asm volatile("s_wait_asynccnt 0x0" ::: "memory");
}
__device__ __forceinline__ unsigned lds_addr(const void* p)
{
    return (unsigned)(size_t)p;   // LDS aperture: addr[31:0] == group offset
}

// ---------------------------------------------------------------------------
// Kernel A: 1x1 conv  y64[b,o,h,w] = sum_c x[b,c,h,w]*w[o,c] + bias[o]
// ---------------------------------------------------------------------------
__global__ __launch_bounds__(256) void conv1x1_kernel(
    const float* __restrict__ x, const float* __restrict__ w,
    const float* __restrict__ bias, float* __restrict__ y64)
{
    __shared__ float wsh[CDIM * 256];
    for (int i = threadIdx.x; i < CDIM * 256; i += 256) wsh[i] = w[i];
    __syncthreads();

    int idx = blockIdx.x * 256 + threadIdx.x;      // 4*4096 threads
    int b = idx >> 12;
    int hw = idx & 4095;
    const float* xb = x + (size_t)b * 256 * HW64 + hw;

    float acc[CDIM];
#pragma unroll
    for (int o = 0; o < CDIM; ++o) acc[o] = bias[o];
    for (int c = 0; c < 256; ++c) {
        float xv = __builtin_nontemporal_load(xb + (size_t)c * HW64);
#pragma unroll
        for (int o = 0; o < CDIM; ++o) acc[o] += wsh[o * 256 + c] * xv;
    }
#pragma unroll
    for (int o = 0; o < CDIM; ++o)
        y64[((size_t)b * CDIM + o) * HW64 + hw] = acc[o];
}

// bilinear taps, align_corners=True: pos = s * 63/511
__device__ __forceinline__ void bilin_taps(int s, int& lo, int& hi, float& fr)
{
    float p = (float)s * (63.0f / 511.0f);
    lo = (int)p;
    fr = p - (float)lo;
    hi = lo + 1; if (hi > 63) hi = 63;
}

// ---------------------------------------------------------------------------
// Kernel B: argmax one-hot + 4x4 mean pool + PxP unfold (y recomputed from
// L2-resident y64). One thread per fm location (b, mh, mw) in [4,128,128].
// ---------------------------------------------------------------------------
__global__ __launch_bounds__(256) void pool_unfold_kernel(
    const float* __restrict__ y64, float* __restrict__ unf)
{
    int idx = blockIdx.x * 256 + threadIdx.x;      // 65536 threads
    int b = idx >> 14;
    int rem = idx & 16383;
    int mh = rem >> 7, mw = rem & 127;

    const float* yb = y64 + (size_t)b * CDIM * HW64;
    float cnt[CDIM] = {};

#pragma unroll
    for (int di = 0; di < 4; ++di) {
        int s = mh * 4 + di;
        int h0, h1; float fh; bilin_taps(s, h0, h1, fh);
#pragma unroll
        for (int dj = 0; dj < 4; ++dj) {
            int t = mw * 4 + dj;
            int w0, w1; float fw; bilin_taps(t, w0, w1, fw);
            int i00 = h0 * 64 + w0, i01 = h0 * 64 + w1;
            int i10 = h1 * 64 + w0, i11 = h1 * 64 + w1;
            float c00 = (1.f - fh) * (1.f - fw), c01 = (1.f - fh) * fw;
            float c10 = fh * (1.f - fw),         c11 = fh * fw;
            int best = 0; float bv = -3.4e38f;
#pragma unroll
            for (int o = 0; o < CDIM; ++o) {
                const float* yp = yb + (size_t)o * HW64;
                float v = c00 * yp[i00] + c01 * yp[i01]
                        + c10 * yp[i10] + c11 * yp[i11];
                if (v > bv) { bv = v; best = o; }   // first-max tiebreak
            }
            cnt[best] += 1.0f;
        }
    }
    int p = (mh & 7) * 8 + (mw & 7);
    int l = (mh >> 3) * 16 + (mw >> 3);
#pragma unroll
    for (int o = 0; o < CDIM; ++o)
        unf[(((size_t)b * CDIM + o) * PSQ + p) * KDIM + l] = cnt[o] * (1.0f / 16.0f);
}

// ---------------------------------------------------------------------------
// Kernel C: per-(b,c) GEMM  corr[q,p] = (1/nz[q]) * sum_k attn[q,k]*unf[p,k]
// M=4096 N=64 K=256 x32 via V_WMMA_F32_16X16X4_F32.
// 256 thr / 8 waves, tile 128x64. K chunks of 32 double-buffered in LDS via
// GLOBAL_LOAD_ASYNC_TO_LDS_B128 (DMA overlaps WMMA). nz count + verbatim attn
// copy (tuple output 2) are serviced from the LDS tile (attn read from DRAM
// exactly once, NT). corr written pre-folded into [b,c,s,t].
// ---------------------------------------------------------------------------
__global__ __launch_bounds__(256) void gemm_corr_kernel(
    const float* __restrict__ attn,      // [4,8,4096,256]
    const float* __restrict__ unf,       // [4,8,64,256]
    float* __restrict__ corrFold,        // d_out[0 : 4*8*512*512]
    float* __restrict__ attnCopy)        // d_out second tuple element
{
    __shared__ float As[2][128 * LDA];   // 2 x 18 KB
    __shared__ float Bs[2][PSQ * LDA];   // 2 x  9 KB
    __shared__ float nzc[128];

    const int mBlk = blockIdx.x;         // 0..31  (128 rows each)
    const int c = blockIdx.y;            // 0..7
    const int b = blockIdx.z;            // 0..3
    const int bc = b * CDIM + c;
    const int tid = threadIdx.x;
    const int lane = tid & 31;
    const int wave = tid >> 5;

    if (tid < 128) nzc[tid] = 0.0f;

    const float* Abase = attn + ((size_t)bc * QDIM + (size_t)mBlk * 128) * KDIM;
    float* Acopy = attnCopy + ((size_t)bc * QDIM + (size_t)mBlk * 128) * KDIM;
    const float* Bbase = unf + (size_t)bc * PSQ * KDIM;

    v8f acc[4] = {};                     // 4 N-tiles of 16x16 f32

    const int rA = tid >> 1;             // 0..127 (A row this thread stages)
    const int cA = (tid & 1) * 16;       // col base, 16 floats
    const int pB = tid >> 2;             // 0..63  (B row this thread stages)
    const int cB = (tid & 3) * 8;        // col base, 8 floats

    const int rBase = wave * 16;
    const int half = lane >> 4;          // A16x4 ISA layout: lanes16-31 = K2,K3
    const int l16 = lane & 15;

    const unsigned ldsA[2] = { lds_addr(&As[0][rA * LDA + cA]),
                               lds_addr(&As[1][rA * LDA + cA]) };
    const unsigned ldsB[2] = { lds_addr(&Bs[0][pB * LDA + cB]),
                               lds_addr(&Bs[1][pB * LDA + cB]) };

    auto issue_chunk = [&](int kc, int buf) {
        const float* gA = Abase + (size_t)rA * KDIM + kc * 32 + cA;
#pragma unroll
        for (int i = 0; i < 4; ++i)
            async_ld16_nt(ldsA[buf] + i * 16, gA + i * 4);   // attn: NT stream
        const float* gB = Bbase + (size_t)pB * KDIM + kc * 32 + cB;
        async_ld16(ldsB[buf], gB);                           // unf: keep in L2
        async_ld16(ldsB[buf] + 16, gB + 4);
    };

    __syncthreads();          // nzc init visible; LDS buffers free
    issue_chunk(0, 0);

    for (int kc = 0; kc < 8; ++kc) {
        const int cur = kc & 1;
        if (kc < 7) { issue_chunk(kc + 1, cur ^ 1); wait_async6(); }
        else        { wait_async0(); }
        __syncthreads();      // chunk kc resident in buf[cur] for all waves

        // ---- nz count + verbatim attn copy, serviced from LDS ----
        {
            float* Ac = Acopy + (size_t)rA * KDIM + kc * 32 + cA;
            const float* ls = &As[cur][rA * LDA + cA];
            float nzl = 0.0f;
#pragma unroll
            for (int i = 0; i < 4; ++i) {
                v4f v = *(const v4f*)(ls + i * 4);
                __builtin_nontemporal_store(v, (v4f*)(Ac + i * 4));
                nzl += (v.x != 0.0f ? 1.0f : 0.0f) + (v.y != 0.0f ? 1.0f : 0.0f)
                     + (v.z != 0.0f ? 1.0f : 0.0f) + (v.w != 0.0f ? 1.0f : 0.0f);
            }
            atomicAdd(&nzc[rA], nzl);    // ds_add_f32
        }

        // ---- 8 k-steps of V_WMMA_F32_16X16X4_F32 per N-tile ----
        const float* Ap = &As[cur][0];
        const float* Bp = &Bs[cur][0];
#pragma unroll
        for (int kt = 0; kt < 8; ++kt) {
            const int k = kt * 4 + 2 * half;
            v2f a;
            a.x = Ap[(rBase + l16) * LDA + k];
            a.y = Ap[(rBase + l16) * LDA + k + 1];
#pragma unroll
            for (int nt = 0; nt < 4; ++nt) {
                v2f bf;
                bf.x = Bp[(nt * 16 + l16) * LDA + k];
                bf.y = Bp[(nt * 16 + l16) * LDA + k + 1];
                acc[nt] = __builtin_amdgcn_wmma_f32_16x16x4_f32(
                    false, a, false, bf, (short)0, acc[nt], false, false);
            }
        }
        __syncthreads();      // all waves done reading buf[cur]
    }

    // ---- epilogue: 1/nz scale + fold (q,p)->(s,t) and store ----
#pragma unroll
    for (int v = 0; v < 8; ++v) {
        int m = rBase + v + 8 * half;              // C/D layout: M = v + 8*group
        float scale = 1.0f / (nzc[m] + 1e-5f);
        int q = mBlk * 128 + m;
        int qh = q >> 6, qw = q & 63;
#pragma unroll
        for (int nt = 0; nt < 4; ++nt) {
            int p = nt * 16 + l16;                 // C/D layout: N = lane%16
            int s = qh * 8 + (p >> 3);
            int t = qw * 8 + (p & 7);
            corrFold[((size_t)bc * SDIM + s) * SDIM + t] = acc[nt][v] * scale;
        }
    }
}

// ---------------------------------------------------------------------------
// Kernel D: out = log_softmax((corr+1)*y, axis=c), y recomputed bilinearly
// from y64. In-place on d_out's corr region (each thread RMWs only its own
// 8 addresses -> race-free; corr is L2-hot from kernel C).
// ---------------------------------------------------------------------------
__global__ __launch_bounds__(256) void softmax_kernel(
    const float* __restrict__ y64, float* __restrict__ out)
{
    int idx = blockIdx.x * 256 + threadIdx.x;      // 4*512*512 threads
    int b = idx >> 18;
    int sp = idx & 262143;
    int s = sp >> 9, t = sp & 511;

    int h0, h1; float fh; bilin_taps(s, h0, h1, fh);
    int w0, w1; float fw; bilin_taps(t, w0, w1, fw);
    int i00 = h0 * 64 + w0, i01 = h0 * 64 + w1;
    int i10 = h1 * 64 + w0, i11 = h1 * 64 + w1;
    float c00 = (1.f - fh) * (1.f - fw), c01 = (1.f - fh) * fw;
    float c10 = fh * (1.f - fw),         c11 = fh * fw;

    const float* yb = y64 + (size_t)b * CDIM * HW64;
    size_t obase = ((size_t)b * CDIM * SDIM + s) * SDIM + t;

    float v[CDIM];
    float mx = -3.4e38f;
#pragma unroll
    for (int o = 0; o < CDIM; ++o) {
        const float* yp = yb + (size_t)o * HW64;
        float yv = c00 * yp[i00] + c01 * yp[i01] + c10 * yp[i10] + c11 * yp[i11];
        float corr = out[obase + (size_t)o * SDIM * SDIM];
        v[o] = (corr + 1.0f) * yv;
        mx = fmaxf(mx, v[o]);
    }
    float sum = 0.0f;
#pragma unroll
    for (int o = 0; o < CDIM; ++o) sum += expf(v[o] - mx);
    float ls = mx + logf(sum);
#pragma unroll
    for (int o = 0; o < CDIM; ++o)
        out[obase + (size_t)o * SDIM * SDIM] = v[o] - ls;
}

// ---------------------------------------------------------------------------
extern "C" void kernel_launch(void* const* d_in, const int* in_sizes, int n_in,
                              void* d_out, int out_size, void* d_ws, size_t ws_size,
                              hipStream_t stream)
{
    const float* x      = (const float*)d_in[0];   // [4,256,64,64]
    const float* attn   = (const float*)d_in[1];   // [4,8,4096,256]
    const float* conv_w = (const float*)d_in[2];   // [8,256]
    const float* conv_b = (const float*)d_in[3];   // [8]

    float* out      = (float*)d_out;                              // [4,8,512,512]
    float* attnCopy = out + (size_t)BDIM * CDIM * SDIM * SDIM;    // tuple elem 2

    float* y64 = (float*)d_ws;                       // 4*8*64*64   (512 KB)
    float* unf = y64 + (size_t)BDIM * CDIM * HW64;   // 4*8*64*256  (2 MB)

    conv1x1_kernel<<<64, 256, 0, stream>>>(x, conv_w, conv_b, y64);
    pool_unfold_kernel<<<256, 256, 0, stream>>>(y64, unf);
    dim3 g(32, CDIM, BDIM);
    gemm_corr_kernel<<<g, 256, 0, stream>>>(attn, unf, out, attnCopy);
    softmax_kernel<<<4096, 256, 0, stream>>>(y64, out);
}